// FilterModel_v2_25237227831812
// MI455X (gfx1250) — compile-verified
//
#include <hip/hip_runtime.h>
#include <hip/hip_bf16.h>
#include <stdint.h>

// Problem constants from the reference: B=100, R=262144=2^18, K=32.
#define TPB     256
#define KMATCH  32
#define RLOG2   18
#define RMASK   ((1 << RLOG2) - 1)
#define DEPTH   4            // async pipeline stages (16KB LDS per block)

typedef __attribute__((ext_vector_type(4))) float f4;

// ---- CDNA5 async global->LDS copy (ASYNCcnt domain, CDNA5 ISA) ----
// Per-lane 16B DMA: VDST = 32-bit LDS byte offset, VADDR = 64-bit global address.
__device__ __forceinline__ void async_load_b128(unsigned lds_off, unsigned long long gaddr) {
  asm volatile("global_load_async_to_lds_b128 %0, %1, off"
               :: "v"(lds_off), "v"(gaddr)
               : "memory");
}
// Immediate-operand waits: pick the literal that retires the oldest outstanding load.
__device__ __forceinline__ void wait_async_le(int inflight) {
  switch (inflight) {
    case 1:  asm volatile("s_wait_asynccnt 0x0" ::: "memory"); break;
    case 2:  asm volatile("s_wait_asynccnt 0x1" ::: "memory"); break;
    case 3:  asm volatile("s_wait_asynccnt 0x2" ::: "memory"); break;
    default: asm volatile("s_wait_asynccnt 0x3" ::: "memory"); break;
  }
}
__device__ __forceinline__ void wait_ds0() {
  asm volatile("s_wait_dscnt 0x0" ::: "memory");
}

// Kernel 1: zero the per-row match counters (d_ws is poisoned, not re-zeroed by harness).
__global__ void zero_counts_kernel(int* __restrict__ cnt, int n) {
  int i = blockIdx.x * blockDim.x + threadIdx.x;
  if (i < n) cnt[i] = 0;
}

// Kernel 2: HBM-rate stream with a 4-deep async global->LDS pipeline (2KB in flight
// per wave). Compare against t, NT-store the 0/1 output, record rare matches.
__global__ void __launch_bounds__(TPB) filter_stream_kernel(
    const float* __restrict__ in, const int* __restrict__ target,
    float* __restrict__ out_diff, int* __restrict__ cnt, int* __restrict__ slots,
    long long n4) {
  __shared__ f4 stage[DEPTH][TPB];
  const int tid = threadIdx.x;
  const float tf = (float)(*target + 1);   // == 8.0f

  const long long i0 = (long long)blockIdx.x * TPB + tid;
  const long long stride = (long long)gridDim.x * TPB;

  // Prime the pipeline: up to DEPTH-1 loads in flight before the first consume.
  int inflight = 0;
#pragma unroll
  for (int p = 0; p < DEPTH - 1; ++p) {
    const long long ip = i0 + (long long)p * stride;
    if (ip < n4) {
      async_load_b128((unsigned)(uintptr_t)(void*)&stage[p][tid],
                      (unsigned long long)(uintptr_t)(in + 4 * ip));
      ++inflight;
    }
  }

  int st = 0;
  for (long long i = i0; i < n4; i += stride) {
    // Issue stage i+ (DEPTH-1) strides ahead, then retire the oldest.
    const long long nx = i + (long long)(DEPTH - 1) * stride;
    if (nx < n4) {
      async_load_b128((unsigned)(uintptr_t)(void*)&stage[(st + DEPTH - 1) % DEPTH][tid],
                      (unsigned long long)(uintptr_t)(in + 4 * nx));
      ++inflight;
    }
    wait_async_le(inflight);                 // oldest async load (this stage) complete
    f4 v = stage[st][tid];
    wait_ds0();                              // LDS read retired before this buffer is re-targeted
    --inflight;

    const long long g = 4 * i;               // flat float index; 4 elems stay in one row (R%4==0)
    const int b = (int)(g >> RLOG2);
    const int r = (int)(g & RMASK);

    f4 o;
    o.x = (v.x == tf) ? 1.0f : 0.0f;
    o.y = (v.y == tf) ? 1.0f : 0.0f;
    o.z = (v.z == tf) ? 1.0f : 0.0f;
    o.w = (v.w == tf) ? 1.0f : 0.0f;

    // Rare path: 3200 matches out of 26.2M elements.
    if (v.x == tf) { int s0 = atomicAdd(cnt + b, 1); if (s0 < KMATCH) slots[b * KMATCH + s0] = r + 0; }
    if (v.y == tf) { int s1 = atomicAdd(cnt + b, 1); if (s1 < KMATCH) slots[b * KMATCH + s1] = r + 1; }
    if (v.z == tf) { int s2 = atomicAdd(cnt + b, 1); if (s2 < KMATCH) slots[b * KMATCH + s2] = r + 2; }
    if (v.w == tf) { int s3 = atomicAdd(cnt + b, 1); if (s3 < KMATCH) slots[b * KMATCH + s3] = r + 3; }

    // Streaming output (210MB total traffic > 192MB L2): non-temporal store.
    __builtin_nontemporal_store(o, (f4*)(out_diff + g));

    st = (st + 1) % DEPTH;
  }
}

// Kernel 3: sort each row's 32 matched indices. One wave32 per row: lane k holds one
// index; rank = #indices smaller (all distinct) via 32 shuffles; scatter as float.
__global__ void order_rows_kernel(const int* __restrict__ slots, float* __restrict__ out_rows) {
  const int b = blockIdx.x;
  const int lane = threadIdx.x;          // 0..31
  int idx = slots[b * KMATCH + lane];
  int rank = 0;
#pragma unroll
  for (int j = 0; j < KMATCH; ++j) {
    int other = __shfl(idx, j, KMATCH);
    rank += (other < idx) ? 1 : 0;
  }
  out_rows[b * KMATCH + rank] = (float)idx;
}

extern "C" void kernel_launch(void* const* d_in, const int* in_sizes, int n_in,
                              void* d_out, int out_size, void* d_ws, size_t ws_size,
                              hipStream_t stream) {
  (void)n_in; (void)out_size; (void)ws_size;

  const float* in  = (const float*)d_in[0];   // block_id, B*R floats (trailing dim 1)
  const int*   tgt = (const int*)d_in[1];     // target_id scalar
  float* out = (float*)d_out;

  const long long BR = (long long)in_sizes[0];     // B*R = 26,214,400
  const int B = (int)(BR >> RLOG2);                // 100
  const long long n4 = BR >> 2;                    // float4 count

  // Workspace layout: [0,512): int cnt[B]; [512, ...): int slots[B*KMATCH]  (~13.3 KB)
  int* cnt   = (int*)d_ws;
  int* slots = (int*)((char*)d_ws + 512);
  float* out_rows = out + BR;                      // tuple outputs concatenated

  zero_counts_kernel<<<(B + 127) / 128, 128, 0, stream>>>(cnt, B);

  // ~8 float4 iterations per thread -> 3200 blocks for this problem size:
  // keeps the 4-deep pipeline primed while leaving abundant wave parallelism.
  const long long iters = 8;
  long long want_threads = (n4 + iters - 1) / iters;
  int grid = (int)((want_threads + TPB - 1) / TPB);
  if (grid < 1) grid = 1;
  filter_stream_kernel<<<grid, TPB, 0, stream>>>(in, tgt, out, cnt, slots, n4);

  order_rows_kernel<<<B, KMATCH, 0, stream>>>(slots, out_rows);
}